// WCE_label_correction_un_4552665333887
// MI455X (gfx1250) — compile-verified
//
#include <hip/hip_runtime.h>

typedef __attribute__((ext_vector_type(4))) float f4;
typedef __attribute__((ext_vector_type(2))) float v2f;
typedef __attribute__((ext_vector_type(8))) float v8f;

#define EPSV 1e-5f
#define LN2F 0.6931471805599453f   // temp = ln2 * (t*log2(pre1+eps) + (1-t)*log2(pre0+eps))

constexpr int Bb      = 64;
constexpr int Hh      = 512;
constexpr int Ww      = 512;
constexpr int NTOT    = Bb * Hh * Ww;        // 16,777,216
constexpr int NVEC    = NTOT / 4;            // 4,194,304 float4's
constexpr int PLANE_V = (Hh * Ww) / 4;       // 65,536 float4's per (H,W) plane
constexpr int NBLK    = 4096;
constexpr int NTHR    = 256;                 // 8 wave32 per block
constexpr int ITERS   = NVEC / (NBLK * NTHR); // 4
static_assert(ITERS * NBLK * NTHR == NVEC, "exact coverage required (EXEC all-ones for WMMA)");
static_assert(PLANE_V == 65536, "shift-based index math assumes 65536");

// Full wave32 reduction of 64 values (2 per lane) using the matrix engine.
// A(16x4) per documented layout: lane L<16 holds A[L][0..1], lane L>=16 holds
// A[L-16][2..3]. With B = all-ones (layout-independent!), D[m][n] = sum_k A[m][k].
// Summing the 8 D VGPRs gives lanes 0-15 = sum(rows 0..7), lanes 16-31 =
// sum(rows 8..15); one xor-16 shuffle finishes. Result broadcast to all lanes.
__device__ __forceinline__ float wave_sum64(float a0, float a1) {
  v2f A;  A[0] = a0;   A[1] = a1;
  v2f Bm; Bm[0] = 1.0f; Bm[1] = 1.0f;
  v8f C = {};
  v8f D = __builtin_amdgcn_wmma_f32_16x16x4_f32(false, A, false, Bm,
                                                (short)0, C, false, false);
  float s = ((D[0] + D[1]) + (D[2] + D[3])) + ((D[4] + D[5]) + (D[6] + D[7]));
  s += __shfl_xor(s, 16, 32);
  return s;
}

__global__ __launch_bounds__(NTHR) void wce_main(
    const float* __restrict__ pred, const float* __restrict__ targ,
    const float* __restrict__ T, const float* __restrict__ mask,
    const float* __restrict__ sigma, float* __restrict__ out,
    float* __restrict__ ws) {
  const f4* p4 = (const f4*)pred;
  const f4* t4 = (const f4*)targ;
  const f4* T4 = (const f4*)T;
  const f4* m4 = (const f4*)mask;
  const f4* s4 = (const f4*)sigma;
  float* outp = out + 1;  // pre1 starts after the scalar loss slot

  const int tid = blockIdx.x * NTHR + threadIdx.x;

  f4 accn = {0.0f, 0.0f, 0.0f, 0.0f};  // accumulates in log2 domain (scaled by ln2 at finalize)
  f4 accd = {0.0f, 0.0f, 0.0f, 0.0f};

#pragma unroll
  for (int it = 0; it < ITERS; ++it) {
    const int i   = tid + it * (NBLK * NTHR);       // vec4 index, in-bounds by construction
    const int b   = i >> 16;                        // i / PLANE_V
    const int r   = i & (PLANE_V - 1);
    const int t0i = b * (2 * PLANE_V) + r;          // T[b,0,...] vec4 index

    // Streaming (read-once) data: non-temporal to keep it out of L2's way.
    f4 p  = __builtin_nontemporal_load(p4 + i);
    f4 tg = __builtin_nontemporal_load(t4 + i);
    f4 T0 = __builtin_nontemporal_load(T4 + t0i);
    f4 T1 = __builtin_nontemporal_load(T4 + t0i + PLANE_V);
    f4 mk = __builtin_nontemporal_load(m4 + i);
    f4 sg = __builtin_nontemporal_load(s4 + i);

#pragma unroll
    for (int j = 0; j < 4; ++j) {
      const float pj  = p[j];
      const float tj  = tg[j];
      const float omp = 1.0f - pj;
      float pre0 = T0[j] * omp + (1.0f - T1[j]) * pj;
      float pre1 = (1.0f - T0[j]) * omp + T1[j] * pj;
      pre0 = fminf(fmaxf(pre0, 0.0f), 1.0f);
      pre1 = fminf(fmaxf(pre1, 0.0f), 1.0f);
      // d_out+1 is only 4B-aligned -> scalar b32 NT stores (still coalesced).
      __builtin_nontemporal_store(pre1, outp + 4l * i + j);
      // log(x) = ln2 * log2(x); ln2 factored out of the whole reduction.
      // Arguments are in [1e-5, 1+1e-5]: v_log_f32 is safe and ~1 ulp here.
      const float temp2 = tj * __log2f(pre1 + EPSV) + (1.0f - tj) * __log2f(pre0 + EPSV);
      const float w     = mk[j] * __expf(-sg[j]);   // mask / exp(sigma)
      accn[j] += temp2 * w;
      accd[j] += mk[j];
    }
  }

  // Wave-level reduction via the matrix engine (EXEC is all-ones here).
  const float sn = wave_sum64(accn[0] + accn[1], accn[2] + accn[3]);
  const float sd = wave_sum64(accd[0] + accd[1], accd[2] + accd[3]);

  __shared__ float s_n[NTHR / 32];
  __shared__ float s_d[NTHR / 32];
  const int lane = threadIdx.x & 31;
  const int wv   = threadIdx.x >> 5;
  if (lane == 0) { s_n[wv] = sn; s_d[wv] = sd; }
  __syncthreads();
  if (threadIdx.x == 0) {
    float tn = 0.0f, td = 0.0f;
#pragma unroll
    for (int w2 = 0; w2 < NTHR / 32; ++w2) { tn += s_n[w2]; td += s_d[w2]; }
    ws[blockIdx.x]        = tn;   // deterministic: plain stores, no atomics
    ws[NBLK + blockIdx.x] = td;
  }
}

__global__ __launch_bounds__(NTHR) void wce_final(const float* __restrict__ ws,
                                                  float* __restrict__ out) {
  float sn = 0.0f, sd = 0.0f;
  // NBLK % NTHR == 0 -> uniform trip count, EXEC stays all-ones for WMMA.
  for (int i = threadIdx.x; i < NBLK; i += NTHR) {
    sn += ws[i];
    sd += ws[NBLK + i];
  }
  const float tn = wave_sum64(sn, 0.0f);
  const float td = wave_sum64(sd, 0.0f);

  __shared__ float s_n[NTHR / 32];
  __shared__ float s_d[NTHR / 32];
  const int lane = threadIdx.x & 31;
  const int wv   = threadIdx.x >> 5;
  if (lane == 0) { s_n[wv] = tn; s_d[wv] = td; }
  __syncthreads();
  if (threadIdx.x == 0) {
    float a = 0.0f, b = 0.0f;
#pragma unroll
    for (int w2 = 0; w2 < NTHR / 32; ++w2) { a += s_n[w2]; b += s_d[w2]; }
    out[0] = -(LN2F * a / b);   // re-apply the hoisted ln2 scale
  }
}

extern "C" void kernel_launch(void* const* d_in, const int* in_sizes, int n_in,
                              void* d_out, int out_size, void* d_ws, size_t ws_size,
                              hipStream_t stream) {
  (void)in_sizes; (void)n_in; (void)out_size; (void)ws_size;
  const float* pred    = (const float*)d_in[0];
  const float* targets = (const float*)d_in[1];
  const float* T       = (const float*)d_in[2];
  const float* mask    = (const float*)d_in[3];
  const float* sigma   = (const float*)d_in[4];
  float* out = (float*)d_out;
  float* ws  = (float*)d_ws;  // needs 2*NBLK*4 = 32 KB

  wce_main<<<NBLK, NTHR, 0, stream>>>(pred, targets, T, mask, sigma, out, ws);
  wce_final<<<1, NTHR, 0, stream>>>(ws, out);
}